// GCNEncoder_66125316489523
// MI455X (gfx1250) — compile-verified
//
#include <hip/hip_runtime.h>
#include <hip/hip_bf16.h>

typedef float v2f __attribute__((ext_vector_type(2)));
typedef float v8f __attribute__((ext_vector_type(8)));

// ---------------------------------------------------------------------------
// Degree / normalization kernels
// ---------------------------------------------------------------------------
__global__ void k_init_deg(float* __restrict__ deg, int n) {
    int i = blockIdx.x * blockDim.x + threadIdx.x;
    if (i < n) deg[i] = 1.0f;                    // self-loop contributes 1
}

__global__ void k_deg_accum(const int* __restrict__ dst, float* __restrict__ deg, int e) {
    int i = blockIdx.x * blockDim.x + threadIdx.x;
    if (i < e) atomicAdd(&deg[dst[i]], 1.0f);
}

__global__ void k_dis(const float* __restrict__ deg, float* __restrict__ dis, int n) {
    int i = blockIdx.x * blockDim.x + threadIdx.x;
    if (i < n) dis[i] = rsqrtf(deg[i]);          // deg >= 1 always (self-loops)
}

// ---------------------------------------------------------------------------
// Dense GEMM  out[M,F] = A[M,K] @ W[K,F]  using V_WMMA_F32_16X16X4_F32.
// K and F are compile-time so all load offsets become 24-bit immediates off a
// single base per operand -> long global_load clauses, minimal address ALU.
// One wave per 16x16 output tile.
// A-operand (f32 16x4): lanes 0-15 hold K=0,1 in v0,v1; lanes 16-31 K=2,3.
// B-operand (f32 4x16): mirrored, N striped across lanes.
// C/D (f32 16x16): 8 VGPRs, lanes 0-15 -> M=r, lanes 16-31 -> M=r+8, N=lane%16.
// ---------------------------------------------------------------------------
template <int K, int F>
__global__ __launch_bounds__(256)
void k_gemm_wmma(const float* __restrict__ A, const float* __restrict__ W,
                 float* __restrict__ out, int M) {
    int wave = (int)((blockIdx.x * blockDim.x + threadIdx.x) >> 5);
    int lane = threadIdx.x & 31;
    constexpr int tilesN = F >> 4;
    int tileM = wave / tilesN;
    int tileN = wave - tileM * tilesN;
    if (tileM * 16 >= M) return;                 // wave-uniform: EXEC stays all-1s

    int half = lane >> 4;                        // 0: K pair {0,1}, 1: K pair {2,3}
    int m    = lane & 15;

    const float* arow = A + (size_t)(tileM * 16 + m) * K + 2 * half;
    const float* wcol = W + (size_t)(2 * half) * F + tileN * 16 + m;

    v8f c = {};
#pragma unroll
    for (int kk = 0; kk < K; kk += 4) {
        v2f a, b;
        a.x = arow[kk + 0];                      // A[row][kk + 2*half + 0]
        a.y = arow[kk + 1];                      // A[row][kk + 2*half + 1]
        b.x = wcol[(kk + 0) * F];                // W[kk + 2*half + 0][col]
        b.y = wcol[(kk + 1) * F];                // W[kk + 2*half + 1][col]
        // (neg_a, A, neg_b, B, c_mod, C, reuse_a, reuse_b)
        c = __builtin_amdgcn_wmma_f32_16x16x4_f32(false, a, false, b,
                                                  (short)0, c, false, false);
    }

    float* orow = out + (size_t)(tileM * 16 + half * 8) * F + tileN * 16 + m;
#pragma unroll
    for (int r = 0; r < 8; ++r) orow[r * F] = c[r];
}

// ---------------------------------------------------------------------------
// agg[n,f] = bias[f] + dis[n]^2 * h[n,f]   (self-loop edge folded in here)
// ---------------------------------------------------------------------------
template <int LOGF>
__global__ void k_init_agg(const float* __restrict__ h, const float* __restrict__ dis,
                           const float* __restrict__ bias, float* __restrict__ agg,
                           int total) {
    int i = blockIdx.x * blockDim.x + threadIdx.x;
    if (i < total) {
        int node = i >> LOGF;
        int f    = i & ((1 << LOGF) - 1);
        float d  = dis[node];
        agg[i] = bias[f] + d * d * h[i];
    }
}

// ---------------------------------------------------------------------------
// Edge scatter: one wave (F=128) / half-wave (F=64) per edge, each lane
// handles 4 consecutive features: coalesced float4 gather of h[src], then
// 4x global_atomic_add_f32 into agg[dst] (L2-resident accumulator).
// LOGC = log2(F/4): 5 for F=128, 4 for F=64.
// ---------------------------------------------------------------------------
template <int LOGC>
__global__ void k_scatter(const float4* __restrict__ h4, const int* __restrict__ src,
                          const int* __restrict__ dst, const float* __restrict__ dis,
                          float* __restrict__ agg, int E) {
    int tid = blockIdx.x * blockDim.x + threadIdx.x;
    int e = tid >> LOGC;
    int c = tid & ((1 << LOGC) - 1);
    if (e >= E) return;
    int s = src[e];
    int d = dst[e];
    float norm = dis[s] * dis[d];
    constexpr int C = 1 << LOGC;                 // float4 chunks per row
    float4 v = h4[(size_t)s * C + c];
    float* o = agg + ((size_t)d * C + c) * 4;
    atomicAdd(o + 0, v.x * norm);
    atomicAdd(o + 1, v.y * norm);
    atomicAdd(o + 2, v.z * norm);
    atomicAdd(o + 3, v.w * norm);
}

__global__ void k_tanh(const float* __restrict__ in, float* __restrict__ out, int total) {
    int i = blockIdx.x * blockDim.x + threadIdx.x;
    if (i < total) out[i] = tanhf(in[i]);
}

// ---------------------------------------------------------------------------
// Orchestration
// ---------------------------------------------------------------------------
extern "C" void kernel_launch(void* const* d_in, const int* in_sizes, int n_in,
                              void* d_out, int out_size, void* d_ws, size_t ws_size,
                              hipStream_t stream) {
    const int IN_CH = 128, HID = 128, OUT_CH = 64;

    const float* x   = (const float*)d_in[0];
    const int*  edge = (const int*)d_in[1];          // [2, E] row-major
    const float* W1  = (const float*)d_in[2];
    const float* b1  = (const float*)d_in[3];
    const float* W2  = (const float*)d_in[4];
    const float* b2  = (const float*)d_in[5];

    const int N = in_sizes[0] / IN_CH;               // 50000
    const int E = in_sizes[1] / 2;                   // 800000
    const int* src = edge;
    const int* dst = edge + E;

    float* ws  = (float*)d_ws;
    float* deg = ws;                                 // [N]
    float* dis = ws + N;                             // [N]
    float* h   = ws + 2 * N;                         // [N, 128] (2N*4 B = 16B-aligned)
    float* agg = h + (size_t)N * HID;                // [N, 128]
    float* t   = agg;                                // layer-2 GEMM reuses agg space [N, 64]
    float* out = (float*)d_out;                      // [N, 64]

    const int TB = 256;

    // --- normalization: deg^{-1/2} with self-loops ---
    k_init_deg <<<(N + TB - 1) / TB, TB, 0, stream>>>(deg, N);
    k_deg_accum<<<(E + TB - 1) / TB, TB, 0, stream>>>(dst, deg, E);
    k_dis      <<<(N + TB - 1) / TB, TB, 0, stream>>>(deg, dis, N);

    // --- layer 1: h = x @ W1 ; agg = b1 + dis^2*h ; scatter ; h = tanh(agg) ---
    {
        int tiles = (N / 16) * (HID / 16);           // 3125 * 8
        k_gemm_wmma<128, 128><<<(tiles + 7) / 8, TB, 0, stream>>>(x, W1, h, N);
    }
    k_init_agg<7><<<(N * HID + TB - 1) / TB, TB, 0, stream>>>(h, dis, b1, agg, N * HID);
    k_scatter<5><<<((size_t)E * 32 + TB - 1) / TB, TB, 0, stream>>>(
        (const float4*)h, src, dst, dis, agg, E);
    k_tanh<<<(N * HID + TB - 1) / TB, TB, 0, stream>>>(agg, h, N * HID);

    // --- layer 2: t = h @ W2 ; out = b2 + dis^2*t ; scatter into out ---
    {
        int tiles = (N / 16) * (OUT_CH / 16);        // 3125 * 4
        k_gemm_wmma<128, 64><<<(tiles + 7) / 8, TB, 0, stream>>>(h, W2, t, N);
    }
    k_init_agg<6><<<(N * OUT_CH + TB - 1) / TB, TB, 0, stream>>>(t, dis, b2, out, N * OUT_CH);
    k_scatter<4><<<((size_t)E * 16 + TB - 1) / TB, TB, 0, stream>>>(
        (const float4*)t, src, dst, dis, out, E);
}